// EdgeConv_81853486727791
// MI455X (gfx1250) — compile-verified
//
#include <hip/hip_runtime.h>

#define B_ 8
#define C_ 64
#define N_ 4096
#define O_ 64
#define KN 20
#define STATS_BLOCKS ((B_ * N_) / 64)

typedef __attribute__((ext_vector_type(2))) float v2f;
typedef __attribute__((ext_vector_type(4))) float v4f;
typedef __attribute__((ext_vector_type(8))) float v8f;

// D(16x16,f32) = A(16x4,f32) * B(4x16,f32) + C  — chained over K
__device__ __forceinline__ v8f wmma4(v2f a, v2f b, v8f c) {
  return __builtin_amdgcn_wmma_f32_16x16x4_f32(false, a, false, b, (short)0, c,
                                               false, false);
}

// -------------------- xt[b,n,c] = x[b,c,n]  (point-major transpose) + sq[b,n]
__global__ __launch_bounds__(256) void prep_kernel(const float* __restrict__ x,
                                                   float* __restrict__ xt,
                                                   float* __restrict__ sq) {
  int g = blockIdx.x * 256 + threadIdx.x;          // 0..B*N-1  (= b*N + n)
  int b = g >> 12;
  int n = g & (N_ - 1);
  const float* xp = x + (size_t)b * C_ * N_ + n;
  float v[C_];
  float s = 0.f;
#pragma unroll
  for (int c = 0; c < C_; ++c) {
    v[c] = xp[(size_t)c * N_];                     // coalesced across threads
    s += v[c] * v[c];
  }
  sq[g] = s;
  v4f* dst = (v4f*)(xt + (size_t)g * C_);          // 256B-aligned row per point
#pragma unroll
  for (int t = 0; t < C_ / 4; ++t) {
    v4f q;
    q[0] = v[4 * t]; q[1] = v[4 * t + 1]; q[2] = v[4 * t + 2]; q[3] = v[4 * t + 3];
    dst[t] = q;
  }
}

// ----------------- distance tiles (WMMA f32, D kept in registers) + per-lane top-k
// A = candidate rows (m), B = this wave's 16 points (n). Lane l owns point
// n0+(l&15); its accumulators hold candidates m = m0 + r + 8*(l>>4) (+16 for acc1).
// K-windows reordered so one b128 load feeds two WMMA steps. An extra WMMA step
// folds -0.5*sq[m] into the accumulator (rank-equivalent to the true pd; -sq[n]
// is per-row constant and only indices are kept). Lane pair lists merged at end.
__global__ __launch_bounds__(32) void topk_kernel(const float* __restrict__ xt,
                                                  const float* __restrict__ sq,
                                                  int* __restrict__ idx_out) {
  __shared__ float svals[16][KN];
  __shared__ int   sinds[16][KN];
  const int lane = threadIdx.x;
  const int b    = blockIdx.y;
  const int n0   = blockIdx.x * 16;                // this wave's 16 points
  const float* xb  = xt + (size_t)b * N_ * C_;
  const float* sqb = sq + (size_t)b * N_;
  const int half = lane >> 4;
  const int lm   = lane & 15;

  // B fragment (fixed): point n0+lm, b128 per pair of K-steps
  v2f bf[16];
#pragma unroll
  for (int t = 0; t < 8; ++t) {
    v4f q = *(const v4f*)(xb + (size_t)(n0 + lm) * C_ + 8 * t + 4 * half);
    bf[2 * t][0]     = q[0]; bf[2 * t][1]     = q[1];
    bf[2 * t + 1][0] = q[2]; bf[2 * t + 1][1] = q[3];
  }
  // extra step: B slot0 = -0.5 (half-0 lanes), pairs with A slot0 = sq[m]
  v2f bfx;
  bfx[0] = half ? 0.f : -0.5f;
  bfx[1] = 0.f;

  float vals[KN];
  int   inds[KN];
#pragma unroll
  for (int s = 0; s < KN; ++s) { vals[s] = -3.4e38f; inds[s] = 0; }
  float kmin = -3.4e38f;
  int   amin = 0;

#define TOPK_UPD(V, M)                                              \
  do {                                                              \
    if ((V) > kmin) {                                               \
      _Pragma("unroll") for (int s_ = 0; s_ < KN; ++s_)             \
          if (s_ == amin) { vals[s_] = (V); inds[s_] = (M); }       \
      kmin = vals[0]; amin = 0;                                     \
      _Pragma("unroll") for (int s_ = 1; s_ < KN; ++s_)             \
          if (vals[s_] < kmin) { kmin = vals[s_]; amin = s_; }      \
    }                                                               \
  } while (0)

  for (int m0 = 0; m0 < N_; m0 += 32) {
    // A fragments: candidate rows m0+lm and m0+16+lm, b128 per 2 steps
    v2f a0[16], a1[16];
#pragma unroll
    for (int t = 0; t < 8; ++t) {
      v4f q0 = *(const v4f*)(xb + (size_t)(m0 + lm) * C_ + 8 * t + 4 * half);
      v4f q1 = *(const v4f*)(xb + (size_t)(m0 + 16 + lm) * C_ + 8 * t + 4 * half);
      a0[2 * t][0]     = q0[0]; a0[2 * t][1]     = q0[1];
      a0[2 * t + 1][0] = q0[2]; a0[2 * t + 1][1] = q0[3];
      a1[2 * t][0]     = q1[0]; a1[2 * t][1]     = q1[1];
      a1[2 * t + 1][0] = q1[2]; a1[2 * t + 1][1] = q1[3];
    }
    float sv0 = sqb[m0 + lm];
    float sv1 = sqb[m0 + 16 + lm];
    v2f a0x, a1x;
    a0x[0] = half ? 0.f : sv0; a0x[1] = 0.f;
    a1x[0] = half ? 0.f : sv1; a1x[1] = 0.f;

    v8f acc0 = {};
    v8f acc1 = {};
#pragma unroll
    for (int s = 0; s < 16; ++s) {
      acc0 = wmma4(a0[s], bf[s], acc0);
      acc1 = wmma4(a1[s], bf[s], acc1);
    }
    acc0 = wmma4(a0x, bfx, acc0);                  // acc = inner - 0.5*sq[m]
    acc1 = wmma4(a1x, bfx, acc1);

    // scan straight from the accumulators: no LDS, no barrier, no epilogue math
#pragma unroll
    for (int r = 0; r < 8; ++r) {
      int m = m0 + r + 8 * half;
      TOPK_UPD(acc0[r], m);
    }
#pragma unroll
    for (int r = 0; r < 8; ++r) {
      int m = m0 + 16 + r + 8 * half;
      TOPK_UPD(acc1[r], m);
    }
  }

  // merge the lane pair's two top-20 lists (once per kernel)
  if (half == 1) {
#pragma unroll
    for (int s = 0; s < KN; ++s) { svals[lm][s] = vals[s]; sinds[lm][s] = inds[s]; }
  }
  __syncthreads();
  if (half == 0) {
#pragma unroll
    for (int s = 0; s < KN; ++s) {
      float v = svals[lm][s];
      int   m = sinds[lm][s];
      TOPK_UPD(v, m);
    }
    int* op = idx_out + ((size_t)b * N_ + n0 + lm) * KN;
#pragma unroll
    for (int s = 0; s < KN; ++s) op[s] = inds[s];
  }
#undef TOPK_UPD
}

// ------------------- proj = xt*w1^T , cterm = xt*(w2-w1)^T  (WMMA, 16 rows per wave)
__global__ __launch_bounds__(32) void proj_kernel(const float* __restrict__ xt,
                                                  const float* __restrict__ w,
                                                  float* __restrict__ proj,
                                                  float* __restrict__ cterm) {
  const int lane = threadIdx.x;
  const int b    = blockIdx.y;
  const int n0   = blockIdx.x * 16;
  const int half = lane >> 4;
  const int lm   = lane & 15;
  const float* xb = xt + (size_t)b * N_ * C_;

  v2f a[16];
#pragma unroll
  for (int t = 0; t < 8; ++t) {
    v4f q = *(const v4f*)(xb + (size_t)(n0 + lm) * C_ + 8 * t + 4 * half);
    a[2 * t][0]     = q[0]; a[2 * t][1]     = q[1];
    a[2 * t + 1][0] = q[2]; a[2 * t + 1][1] = q[3];
  }
#pragma unroll
  for (int o0 = 0; o0 < O_; o0 += 16) {
    v8f accP = {};
    v8f accC = {};
    int o = o0 + lm;
#pragma unroll
    for (int t = 0; t < 8; ++t) {
      v4f q1 = *(const v4f*)(w + o * 128 + 8 * t + 4 * half);        // w1 chunk
      v4f q2 = *(const v4f*)(w + o * 128 + 64 + 8 * t + 4 * half);   // w2 chunk
      v2f b1a, b1b, bda, bdb;
      b1a[0] = q1[0];         b1a[1] = q1[1];
      b1b[0] = q1[2];         b1b[1] = q1[3];
      bda[0] = q2[0] - q1[0]; bda[1] = q2[1] - q1[1];
      bdb[0] = q2[2] - q1[2]; bdb[1] = q2[3] - q1[3];
      accP = wmma4(a[2 * t], b1a, accP);
      accP = wmma4(a[2 * t + 1], b1b, accP);
      accC = wmma4(a[2 * t], bda, accC);
      accC = wmma4(a[2 * t + 1], bdb, accC);
    }
#pragma unroll
    for (int r = 0; r < 8; ++r) {
      int row = n0 + r + 8 * half;
      proj[((size_t)b * N_ + row) * O_ + o]  = accP[r];
      cterm[((size_t)b * N_ + row) * O_ + o] = accC[r];
    }
  }
}

// -------------------------------- per-block deterministic partial sums for batch-norm
__global__ __launch_bounds__(256) void stats_kernel(const int* __restrict__ idx,
                                                    const float* __restrict__ proj,
                                                    const float* __restrict__ cterm,
                                                    float* __restrict__ part) {
  __shared__ float ss[4][64], ss2[4][64];
  int o  = threadIdx.x & 63;
  int rg = threadIdx.x >> 6;
  int base = blockIdx.x * 64;
  float a = 0.f, a2 = 0.f;
  for (int r = rg; r < 64; r += 4) {
    int row  = base + r;                 // = b*N + n
    int boff = row & ~(N_ - 1);          // b*N
    float ct = cterm[(size_t)row * O_ + o];
    const int* ir = idx + (size_t)row * KN;
#pragma unroll
    for (int kk = 0; kk < KN; ++kk) {
      int m   = ir[kk];
      float y = proj[(size_t)(boff + m) * O_ + o] + ct;
      a += y;
      a2 += y * y;
    }
  }
  ss[rg][o]  = a;
  ss2[rg][o] = a2;
  __syncthreads();
  if (rg == 0) {
    part[blockIdx.x * 128 + o]      = ss[0][o] + ss[1][o] + ss[2][o] + ss[3][o];
    part[blockIdx.x * 128 + 64 + o] = ss2[0][o] + ss2[1][o] + ss2[2][o] + ss2[3][o];
  }
}

__global__ __launch_bounds__(64) void finalize_kernel(const float* __restrict__ part,
                                                      const float* __restrict__ gamma,
                                                      const float* __restrict__ beta,
                                                      float* __restrict__ scsh) {
  int o = threadIdx.x;
  float s = 0.f, s2 = 0.f;
  for (int blk = 0; blk < STATS_BLOCKS; ++blk) {   // fixed order -> deterministic
    s  += part[blk * 128 + o];
    s2 += part[blk * 128 + 64 + o];
  }
  const float cnt  = (float)B_ * N_ * KN;
  float mean  = s / cnt;
  float var   = s2 / cnt - mean * mean;
  float scale = rsqrtf(var + 1e-5f) * gamma[o];
  scsh[o]      = scale;
  scsh[64 + o] = beta[o] - mean * scale;
}

// ------------------------- normalize + leaky-relu + max over k + transpose to (B,O,N)
__global__ __launch_bounds__(256) void out_kernel(const int* __restrict__ idx,
                                                  const float* __restrict__ proj,
                                                  const float* __restrict__ cterm,
                                                  const float* __restrict__ scsh,
                                                  float* __restrict__ out) {
  int o  = threadIdx.x & 63;
  int rg = threadIdx.x >> 6;
  int base = blockIdx.x * 64;
  float scale = scsh[o];
  float shift = scsh[64 + o];
  for (int r = rg; r < 64; r += 4) {
    int row  = base + r;
    int boff = row & ~(N_ - 1);
    int b    = row >> 12;
    int n    = row & (N_ - 1);
    float ct = cterm[(size_t)row * O_ + o];
    const int* ir = idx + (size_t)row * KN;
    float mx = -3.4e38f;
#pragma unroll
    for (int kk = 0; kk < KN; ++kk) {
      int m   = ir[kk];
      float y = (proj[(size_t)(boff + m) * O_ + o] + ct) * scale + shift;
      y  = (y >= 0.f) ? y : 0.2f * y;
      mx = fmaxf(mx, y);
    }
    out[((size_t)(b * O_ + o)) * N_ + n] = mx;
  }
}

extern "C" void kernel_launch(void* const* d_in, const int* in_sizes, int n_in,
                              void* d_out, int out_size, void* d_ws, size_t ws_size,
                              hipStream_t stream) {
  (void)in_sizes; (void)n_in; (void)out_size; (void)ws_size;
  const float* x     = (const float*)d_in[0];
  const float* w     = (const float*)d_in[1];
  const float* gamma = (const float*)d_in[2];
  const float* beta  = (const float*)d_in[3];
  // d_in[4] = k (compile-time KN=20)

  char* ws = (char*)d_ws;
  float* sq    = (float*)(ws + 0);          //  131072 B
  int*   idx   = (int*)  (ws + 131072);     // 2621440 B
  float* xt    = (float*)(ws + 2752512);    // 8388608 B  (B,N,C point-major)
  float* proj  = (float*)(ws + 11141120);   // 8388608 B
  float* cterm = (float*)(ws + 19529728);   // 8388608 B
  float* part  = (float*)(ws + 27918336);   //  262144 B
  float* scsh  = (float*)(ws + 28180480);   //     512 B
  float* out   = (float*)d_out;             // (B, O, N) f32

  prep_kernel    <<<dim3((B_ * N_) / 256),     dim3(256), 0, stream>>>(x, xt, sq);
  topk_kernel    <<<dim3(N_ / 16, B_),         dim3(32),  0, stream>>>(xt, sq, idx);
  proj_kernel    <<<dim3(N_ / 16, B_),         dim3(32),  0, stream>>>(xt, w, proj, cterm);
  stats_kernel   <<<dim3(STATS_BLOCKS),        dim3(256), 0, stream>>>(idx, proj, cterm, part);
  finalize_kernel<<<dim3(1),                   dim3(64),  0, stream>>>(part, gamma, beta, scsh);
  out_kernel     <<<dim3(STATS_BLOCKS),        dim3(256), 0, stream>>>(idx, proj, cterm, scsh, out);
}